// CFNO_6330781795048
// MI455X (gfx1250) — compile-verified
//
#include <hip/hip_runtime.h>
#include <hip/hip_bf16.h>

// ---------------------------------------------------------------------------
// CFNO forward, collapsed:
//   y[n,16] = A[16x256] * patch[n,256] + c[16]      (FFT+ComplexLinear+IFFT folded)
//   img[B,16,128,128] -> depthwise 3x3 conv + bias -> BatchNorm(batch stats)
//   -> nearest interp to 128x128 == identity -> out
// ---------------------------------------------------------------------------

typedef __attribute__((ext_vector_type(2))) float v2f;
typedef __attribute__((ext_vector_type(8))) float v8f;

#define LDS_PAD 260          // floats per LDS row: even, 16B-aligned rows, bank-spread
#define TWO_PI 6.283185307179586f

// Async global->LDS (CDNA5 ASYNCcnt path) when the toolchain exposes it.
#if defined(__HIP_DEVICE_COMPILE__) && __has_builtin(__builtin_amdgcn_global_load_async_to_lds_b128)
#define CFNO_ASYNC_LDS 1
typedef int cfno_v4i __attribute__((vector_size(16)));
typedef __attribute__((address_space(1))) cfno_v4i cfno_g_v4i;   // global
typedef __attribute__((address_space(3))) cfno_v4i cfno_l_v4i;   // LDS
#endif

// ---------------- Kernel 1: fold FFT256 -> ComplexLinear -> Re(IFFT16) ------
// A[n,f] = (1/16) sum_j [ (P-S)[n,j]*cos(2pi j f/256) + (Q+R)[n,j]*sin(2pi j f/256) ]
//   P = sum_k cos(2pi k n/16) Wr[k,j]   Q = sum_k cos(..) Wi[k,j]
//   R = sum_k sin(..) Wr[k,j]           S = sum_k sin(..) Wi[k,j]
// c[n] = (1/16) sum_k [ cos(..)(br-bi)_k - sin(..)(br+bi)_k ]
__global__ __launch_bounds__(256) void cfno_precompute(
    const float* __restrict__ Wr, const float* __restrict__ br,
    const float* __restrict__ Wi, const float* __restrict__ bi,
    float* __restrict__ Amat, float* __restrict__ cvec) {
  __shared__ float U[256];
  __shared__ float V[256];
  const int n = blockIdx.x;      // 0..15 (output position of IFFT16)
  const int j = threadIdx.x;     // 0..255 (FFT bin)
  const float w16 = TWO_PI / 16.0f;
  float P = 0.f, Q = 0.f, R = 0.f, S = 0.f;
  for (int k = 0; k < 16; ++k) {
    float ang = (float)((k * n) & 15) * w16;
    float ck = __cosf(ang), sk = __sinf(ang);
    float wr = Wr[k * 256 + j];
    float wi = Wi[k * 256 + j];
    P += ck * wr; Q += ck * wi; R += sk * wr; S += sk * wi;
  }
  U[j] = P - S;
  V[j] = Q + R;
  if (j == 0) {
    float c = 0.f;
    for (int k = 0; k < 16; ++k) {
      float ang = (float)((k * n) & 15) * w16;
      c += __cosf(ang) * (br[k] - bi[k]) - __sinf(ang) * (br[k] + bi[k]);
    }
    cvec[n] = c * (1.0f / 16.0f);
  }
  __syncthreads();
  const int f = j;               // 0..255 (patch-feature index)
  const float w256 = TWO_PI / 256.0f;
  float a = 0.f;
  for (int jj = 0; jj < 256; ++jj) {
    float ang = (float)((jj * f) & 255) * w256;
    a += U[jj] * __cosf(ang) + V[jj] * __sinf(ang);
  }
  Amat[n * 256 + f] = a * (1.0f / 16.0f);
}

// ---------------- Kernel 2: WMMA GEMM  y = p @ A^T + c ----------------------
// 2 waves / block. Each wave: 16 consecutive patches (same image row of patches),
// i.e. a 16-row x 256-col fp32 block of x. K=256 via 64x V_WMMA_F32_16X16X4_F32.
__global__ __launch_bounds__(64) void cfno_gemm(
    const float* __restrict__ x, const float* __restrict__ Amat,
    const float* __restrict__ cvec, float* __restrict__ img) {
  __shared__ __align__(16) float Bs[16 * LDS_PAD];       // transform A (B operand)
  __shared__ __align__(16) float Ps[2][16 * LDS_PAD];    // per-wave patch tiles

  const int t = threadIdx.x;
  const int wave = t >> 5;
  const int lane = t & 31;

  // Tile of 16 consecutive patches.
  const int T = blockIdx.x * 2 + wave;           // 0..8191
  const int n0 = T << 4;                         // first patch id
  const int b   = n0 >> 14;                      // 16384 patches per image
  const int rem = n0 & 16383;
  const int hh  = rem >> 7;                      // patch row (0..127)
  const int ww0 = rem & 127;                     // patch col base (mult of 16)

  const float* gbase = x + ((size_t)b * 2048 + (size_t)hh * 16) * 2048 + (size_t)ww0 * 16;
  float* Pw = &Ps[wave][0];

  // Stage the 16 rows x 256 floats tile into LDS; lane covers 8 floats per row.
#ifdef CFNO_ASYNC_LDS
  for (int s1 = 0; s1 < 16; ++s1) {
    const float* gr = gbase + s1 * 2048 + lane * 8;
    float* lr = Pw + s1 * LDS_PAD + lane * 8;
    __builtin_amdgcn_global_load_async_to_lds_b128(
        (cfno_g_v4i*)gr, (cfno_l_v4i*)lr, 0, 0);
    __builtin_amdgcn_global_load_async_to_lds_b128(
        (cfno_g_v4i*)(gr + 4), (cfno_l_v4i*)(lr + 4), 0, 0);
  }
#else
  // gfx1250 prefetch of the tile (global_prefetch_b8).
  for (int s1 = 0; s1 < 16; ++s1)
    __builtin_prefetch(gbase + s1 * 2048 + lane * 8, 0, 3);
#endif

  // Cooperative load of the 16x256 transform matrix into LDS (float4).
  for (int i = t; i < 1024; i += 64) {          // 1024 float4 = 4096 floats
    int nrow = i >> 6;                           // 64 float4 per row
    int f4 = i & 63;
    *(float4*)&Bs[nrow * LDS_PAD + f4 * 4] = *(const float4*)&Amat[i * 4];
  }

#ifdef CFNO_ASYNC_LDS
#if __has_builtin(__builtin_amdgcn_s_wait_asynccnt)
  __builtin_amdgcn_s_wait_asynccnt(0);
#else
  asm volatile("s_wait_asynccnt 0" ::: "memory");
#endif
#else
  for (int s1 = 0; s1 < 16; ++s1) {
    const float* gr = gbase + s1 * 2048 + lane * 8;
    float* lr = Pw + s1 * LDS_PAD + lane * 8;
    *(float4*)lr       = *(const float4*)gr;
    *(float4*)(lr + 4) = *(const float4*)(gr + 4);
  }
#endif
  __syncthreads();

  // WMMA f32 16x16x4 loop over K=256.
  // A operand (patches): lane<16 -> M=lane,   K = k0+0,k0+1
  //                      lane>=16-> M=lane-16,K = k0+2,k0+3
  // B operand (transform): same K split, N = lane&15; Bs row n holds A[n][0..255].
  v8f acc = {};
  const int m16  = (lane & 15) * 16;        // patch column base inside LDS row
  const int nrow = (lane & 15) * LDS_PAD;   // B row base
  const int fo   = (lane >> 4) * 2;         // K sub-offset per half-wave
  for (int k0 = 0; k0 < 256; k0 += 4) {
    int f0 = k0 + fo;
    v2f av = *(const v2f*)&Pw[((f0 >> 4) * LDS_PAD) + m16 + (f0 & 15)];
    v2f bv = *(const v2f*)&Bs[nrow + f0];
    acc = __builtin_amdgcn_wmma_f32_16x16x4_f32(
        false, av, false, bv, (short)0, acc, false, false);
  }

  // D layout: VGPR r -> M=r (lanes 0-15), M=8+r (lanes 16-31); N = lane&15.
  const int d = lane & 15;
  const float bias = cvec[d];
  const int mo = (lane >> 4) * 8;
  size_t obase = (((size_t)b * 16 + d) * 128 + hh) * 128 + ww0;
  #pragma unroll
  for (int r = 0; r < 8; ++r)
    img[obase + mo + r] = acc[r] + bias;
}

// ---------------- Kernel 3: depthwise 3x3 conv + per-plane partial stats ----
__global__ __launch_bounds__(256) void cfno_conv(
    const float* __restrict__ img, const float* __restrict__ convw,
    const float* __restrict__ convb, float* __restrict__ conved,
    float* __restrict__ part) {
  const int blk = blockIdx.x;           // b*16 + d   (0..127)
  const int d = blk & 15;
  const int t = threadIdx.x;
  float w[9];
  #pragma unroll
  for (int i = 0; i < 9; ++i) w[i] = convw[d * 9 + i];
  const float cb = convb[d];
  const float* plane = img + (size_t)blk * 16384;
  float* oplane = conved + (size_t)blk * 16384;
  float s = 0.f, s2 = 0.f;
  for (int p = t; p < 16384; p += 256) {
    int row = p >> 7, col = p & 127;
    float acc = cb;
    #pragma unroll
    for (int dy = -1; dy <= 1; ++dy) {
      int rr = row + dy;
      if (rr < 0 || rr > 127) continue;
      #pragma unroll
      for (int dx = -1; dx <= 1; ++dx) {
        int cc = col + dx;
        if (cc < 0 || cc > 127) continue;
        acc += w[(dy + 1) * 3 + (dx + 1)] * plane[rr * 128 + cc];
      }
    }
    oplane[p] = acc;
    s += acc;
    s2 += acc * acc;
  }
  __shared__ float rs[256];
  __shared__ float rq[256];
  rs[t] = s; rq[t] = s2;
  __syncthreads();
  for (int off = 128; off > 0; off >>= 1) {
    if (t < off) { rs[t] += rs[t + off]; rq[t] += rq[t + off]; }
    __syncthreads();
  }
  if (t == 0) { part[blk * 2] = rs[0]; part[blk * 2 + 1] = rq[0]; }
}

// ---------------- Kernel 4: finalize BN scale/shift per channel -------------
__global__ void cfno_stats(const float* __restrict__ part,
                           const float* __restrict__ gamma,
                           const float* __restrict__ beta,
                           float* __restrict__ ss) {
  const int d = threadIdx.x;
  if (d >= 16) return;
  float s = 0.f, s2 = 0.f;
  for (int b = 0; b < 8; ++b) {
    s  += part[(b * 16 + d) * 2];
    s2 += part[(b * 16 + d) * 2 + 1];
  }
  const float inv = 1.0f / (8.0f * 128.0f * 128.0f);
  float mean = s * inv;
  float var = s2 * inv - mean * mean;
  float sc = gamma[d] * rsqrtf(var + 1e-5f);
  ss[d] = sc;
  ss[16 + d] = beta[d] - mean * sc;
}

// ---------------- Kernel 5: apply BN (interp is identity) -------------------
__global__ __launch_bounds__(256) void cfno_apply(
    const float* __restrict__ conved, const float* __restrict__ ss,
    float* __restrict__ out) {
  const int i = (blockIdx.x * 256 + threadIdx.x) * 4;
  const int d = (i >> 14) & 15;              // plane = b*16+d, plane size 2^14
  const float sc = ss[d], sh = ss[16 + d];
  float4 v = *(const float4*)(conved + i);
  v.x = v.x * sc + sh;
  v.y = v.y * sc + sh;
  v.z = v.z * sc + sh;
  v.w = v.w * sc + sh;
  *(float4*)(out + i) = v;
}

// ---------------------------------------------------------------------------
extern "C" void kernel_launch(void* const* d_in, const int* in_sizes, int n_in,
                              void* d_out, int out_size, void* d_ws, size_t ws_size,
                              hipStream_t stream) {
  const float* x     = (const float*)d_in[0];   // [8,1,2048,2048]
  const float* W_r   = (const float*)d_in[1];   // [16,256]
  const float* b_r   = (const float*)d_in[2];   // [16]
  const float* W_i   = (const float*)d_in[3];   // [16,256]
  const float* b_i   = (const float*)d_in[4];   // [16]
  const float* convw = (const float*)d_in[5];   // [16,1,3,3]
  const float* convb = (const float*)d_in[6];   // [16]
  const float* gamma = (const float*)d_in[7];   // [16]
  const float* beta  = (const float*)d_in[8];   // [16]
  float* out = (float*)d_out;                   // [8,16,128,128]

  float* ws = (float*)d_ws;
  // workspace layout (floats)
  const size_t A_OFF    = 0;              // 4096
  const size_t C_OFF    = 4096;           // 16
  const size_t SS_OFF   = 4128;           // 32
  const size_t PART_OFF = 4160;           // 256
  const size_t IMG_OFF  = 8192;           // 2,097,152
  const size_t CONV_OFF = IMG_OFF + 2097152;  // 2,097,152

  float* Amat   = ws + A_OFF;
  float* cvec   = ws + C_OFF;
  float* ssbuf  = ws + SS_OFF;
  float* part   = ws + PART_OFF;
  float* img    = ws + IMG_OFF;
  float* conved = ws + CONV_OFF;

  cfno_precompute<<<16, 256, 0, stream>>>(W_r, b_r, W_i, b_i, Amat, cvec);
  cfno_gemm<<<4096, 64, 0, stream>>>(x, Amat, cvec, img);       // 8192 wave-tiles
  cfno_conv<<<128, 256, 0, stream>>>(img, convw, convb, conved, part);
  cfno_stats<<<1, 16, 0, stream>>>(part, gamma, beta, ssbuf);
  cfno_apply<<<2048, 256, 0, stream>>>(conved, ssbuf, out);
}